// IntrinsicReward_77403900609149
// MI455X (gfx1250) — compile-verified
//
#include <hip/hip_runtime.h>
#include <cmath>

typedef __attribute__((ext_vector_type(2))) float v2f;
typedef __attribute__((ext_vector_type(8))) float v8f;
typedef __attribute__((ext_vector_type(4))) float v4f;
typedef __attribute__((ext_vector_type(4))) int   v4i;

#define DEV_INLINE __device__ __forceinline__

DEV_INLINE void atomAddF(float* p, float v) {
    __hip_atomic_fetch_add(p, v, __ATOMIC_RELAXED, __HIP_MEMORY_SCOPE_AGENT);
}

// ---------------- degree / normalization ----------------

__global__ void init_deg_kernel(float* __restrict__ deg, int n) {
    int i = blockIdx.x * blockDim.x + threadIdx.x;
    if (i < n) deg[i] = 1.0f;  // self-loop contributes 1 to every node's degree
}

__global__ void degree4_kernel(const v4i* __restrict__ dst4, float* __restrict__ deg,
                               int E4) {
    int i = blockIdx.x * blockDim.x + threadIdx.x;
    int stride = gridDim.x * blockDim.x;
    for (; i < E4; i += stride) {
        v4i d = __builtin_nontemporal_load(dst4 + i);  // streaming: keep out of L2
        atomAddF(&deg[d.x], 1.0f);
        atomAddF(&deg[d.y], 1.0f);
        atomAddF(&deg[d.z], 1.0f);
        atomAddF(&deg[d.w], 1.0f);
    }
}

__global__ void degree_tail_kernel(const int* __restrict__ dst, float* __restrict__ deg,
                                   int rem) {
    int i = threadIdx.x;
    if (i < rem) atomAddF(&deg[dst[i]], 1.0f);
}

__global__ void dinv_kernel(float* __restrict__ deg, int n) {
    int i = blockIdx.x * blockDim.x + threadIdx.x;
    if (i < n) {
        float d = deg[i];
        deg[i] = (d > 0.0f) ? rsqrtf(fmaxf(d, 1.0f)) : 0.0f;  // in place: deg -> dinv
    }
}

// ---------------- node transform: t = diag(dinv) * (h @ W) via V_WMMA_F32_16X16X4_F32 ----
// Each wave32 transforms 16 nodes. f32 WMMA keeps reference precision; the
// source-side normalization factor dinv[row] is folded into the A operand
// ((dinv*h) @ W == dinv*(h@W) row-wise), so the edge scatter needs no norm
// array at all: agg[dst] += t[src], finalize applies the dinv[dst] factor.
// A layout: lane l -> M = l%16; VGPR v in {0,1} -> K = 2*(l/16) + v.
// B layout: lane l -> N = l%16; VGPR v in {0,1} -> K = 2*(l/16) + v.
// D layout: lane l, VGPR v -> row M = v + 8*(l/16), col N = l%16.
// B padding columns (N=8..15) are discarded at store time, so B loads are
// unconditional from a clamped W address -> no exec-mask divergence.
template <int D>
__global__ void transform_wmma_kernel(const float* __restrict__ h,
                                      const float* __restrict__ W,
                                      const float* __restrict__ dinv,
                                      float* __restrict__ t, int n) {
    int wave = (blockIdx.x * blockDim.x + threadIdx.x) >> 5;
    int lane = threadIdx.x & 31;
    int base = wave * 16;
    int m = lane & 15;
    int half = lane >> 4;

    int row = base + m;
    int rowc = row < n ? row : (n - 1);  // clamp loads; keep EXEC all-1 for WMMA
    const float* hrow = h + (size_t)rowc * D;
    float di = dinv[rowc];  // source-side GCN normalization factor
    int nn = m & 7;  // clamped W column (cols 8..15 read garbage that is discarded)

    v8f c = {};
    constexpr int STEPS = (D + 3) / 4;
#pragma unroll
    for (int s = 0; s < STEPS; ++s) {
        int ka = 4 * s + 2 * half;  // K index of a.x; a.y is ka+1
        v2f a, b;
        if constexpr (D >= 4) {
            // ka, ka+1 always < D: unconditional contiguous loads
            a.x = di * hrow[ka];
            a.y = di * hrow[ka + 1];
            b.x = W[ka * 8 + nn];
            b.y = W[(ka + 1) * 8 + nn];
        } else {
            // D == 2: upper half-lanes (K = 2,3) are zero-padding in A
            float a0 = di * hrow[0];
            float a1 = di * hrow[1];
            a.x = half ? 0.0f : a0;  // branchless cndmask
            a.y = half ? 0.0f : a1;
            int k0 = (ka < D) ? ka : (D - 1);  // clamp W row for padding lanes
            b.x = W[k0 * 8 + nn];
            b.y = W[((k0 + 1 < D) ? k0 + 1 : D - 1) * 8 + nn];
        }
        c = __builtin_amdgcn_wmma_f32_16x16x4_f32(
                /*neg_a=*/false, a, /*neg_b=*/false, b,
                /*c_mod=*/(short)0, c, /*reuse_a=*/false, /*reuse_b=*/false);
    }

    // Store: lane writes column m, rows base + 8*half + v.  Tile validity is
    // wave-uniform -> one scalar branch; fast path uses immediate offsets.
    bool full = (base + 16 <= n);
    if (m < 8) {
        float* p = t + ((size_t)(base + 8 * half)) * 8 + m;
        if (full) {
#pragma unroll
            for (int v = 0; v < 8; ++v) p[v * 8] = c[v];
        } else {
#pragma unroll
            for (int v = 0; v < 8; ++v) {
                if (base + 8 * half + v < n) p[v * 8] = c[v];
            }
        }
    }
}

// ---------------- edge scatter: agg[dst] += t[src] (norm factors pre-folded) ----------------

DEV_INLINE void scatter_one(int s, int d, const float* __restrict__ t,
                            float* __restrict__ agg) {
    const v4f* ts = reinterpret_cast<const v4f*>(t + (size_t)s * 8);
    v4f lo = ts[0];  // gather table: regular-temporal, lives in L2
    v4f hi = ts[1];
    float* ap = agg + (size_t)d * 8;
    atomAddF(ap + 0, lo.x);
    atomAddF(ap + 1, lo.y);
    atomAddF(ap + 2, lo.z);
    atomAddF(ap + 3, lo.w);
    atomAddF(ap + 4, hi.x);
    atomAddF(ap + 5, hi.y);
    atomAddF(ap + 6, hi.z);
    atomAddF(ap + 7, hi.w);
}

__global__ void scatter4_kernel(const v4i* __restrict__ src4, const v4i* __restrict__ dst4,
                                const float* __restrict__ t, float* __restrict__ agg,
                                int E4) {
    int i = blockIdx.x * blockDim.x + threadIdx.x;
    int stride = gridDim.x * blockDim.x;
    for (; i < E4; i += stride) {
        v4i s = __builtin_nontemporal_load(src4 + i);  // streaming reads bypass L2
        v4i d = __builtin_nontemporal_load(dst4 + i);
        scatter_one(s.x, d.x, t, agg);
        scatter_one(s.y, d.y, t, agg);
        scatter_one(s.z, d.z, t, agg);
        scatter_one(s.w, d.w, t, agg);
    }
}

__global__ void scatter_tail_kernel(const int* __restrict__ src, const int* __restrict__ dst,
                                    const float* __restrict__ t, float* __restrict__ agg,
                                    int rem) {
    int i = threadIdx.x;
    if (i < rem) scatter_one(src[i], dst[i], t, agg);
}

// ---------------- finalize: dst-side norm + self-loop + bias + ELU (+ pool) ----------------

__global__ void finalize_kernel(const float* __restrict__ t, const float* __restrict__ agg,
                                const float* __restrict__ dinv, const float* __restrict__ bias,
                                float* __restrict__ hout, float* __restrict__ pooled,
                                int n, int do_pool) {
    __shared__ float ps[8];
    if (threadIdx.x < 8) ps[threadIdx.x] = 0.0f;
    __syncthreads();

    float acc[8] = {0.f, 0.f, 0.f, 0.f, 0.f, 0.f, 0.f, 0.f};
    int i = blockIdx.x * blockDim.x + threadIdx.x;
    int stride = gridDim.x * blockDim.x;
    for (; i < n; i += stride) {
        float di = dinv[i];  // dst-side factor; self-loop term dinv^2*(h@W) = dinv*t
#pragma unroll
        for (int k = 0; k < 8; ++k) {
            float v = di * (agg[(size_t)i * 8 + k] + t[(size_t)i * 8 + k]) + bias[k];
            v = (v > 0.0f) ? v : expm1f(v);  // ELU, alpha = 1
            hout[(size_t)i * 8 + k] = v;
            acc[k] += v;
        }
    }

    if (do_pool) {
#pragma unroll
        for (int k = 0; k < 8; ++k)
            __hip_atomic_fetch_add(&ps[k], acc[k], __ATOMIC_RELAXED,
                                   __HIP_MEMORY_SCOPE_WORKGROUP);
        __syncthreads();
        if (threadIdx.x < 8) atomAddF(&pooled[threadIdx.x], ps[threadIdx.x]);
    }
}

// ---------------- MLP head on pooled mean ----------------

__global__ void head_kernel(const float* __restrict__ pooled,
                            const float* __restrict__ Wr1, const float* __restrict__ br1,
                            const float* __restrict__ Wr2, const float* __restrict__ br2,
                            float* __restrict__ out, int n) {
    if (blockIdx.x == 0 && threadIdx.x == 0) {
        float p[8], q[8];
        float inv_n = 1.0f / (float)n;
#pragma unroll
        for (int i = 0; i < 8; ++i) p[i] = pooled[i] * inv_n;
#pragma unroll
        for (int j = 0; j < 8; ++j) {
            float s = br1[j];
#pragma unroll
            for (int i = 0; i < 8; ++i) s += p[i] * Wr1[i * 8 + j];
            q[j] = (s > 0.0f) ? s : expm1f(s);
        }
        float v = br2[0];
#pragma unroll
        for (int j = 0; j < 8; ++j) v += q[j] * Wr2[j];
        out[0] = v;
    }
}

extern "C" void kernel_launch(void* const* d_in, const int* in_sizes, int n_in,
                              void* d_out, int out_size, void* d_ws, size_t ws_size,
                              hipStream_t stream) {
    const float* x   = (const float*)d_in[0];
    const int*   ei  = (const int*)d_in[1];
    const float* W1  = (const float*)d_in[2];
    const float* b1  = (const float*)d_in[3];
    const float* W2  = (const float*)d_in[4];
    const float* b2  = (const float*)d_in[5];
    const float* W3  = (const float*)d_in[6];
    const float* b3  = (const float*)d_in[7];
    const float* Wr1 = (const float*)d_in[8];
    const float* br1 = (const float*)d_in[9];
    const float* Wr2 = (const float*)d_in[10];
    const float* br2 = (const float*)d_in[11];

    int n = in_sizes[0] / 2;  // NDIM = 2
    int E = in_sizes[1] / 2;  // edge_index is (2, E); 16M fits in int32
    const int* src = ei;
    const int* dst = ei + E;
    int E4  = E >> 2;
    int rem = E - (E4 << 2);
    const v4i* src4 = (const v4i*)src;
    const v4i* dst4 = (const v4i*)dst;
    const int* srcT = src + (E4 << 2);
    const int* dstT = dst + (E4 << 2);

    auto al = [](size_t v) { return (v + 255) & ~(size_t)255; };
    char* w = (char*)d_ws;
    float* deg    = (float*)w; w += al((size_t)n * 4);  // becomes dinv in place
    float* t      = (float*)w; w += al((size_t)n * 8 * 4);
    float* agg    = (float*)w; w += al((size_t)n * 8 * 4);
    float* h      = (float*)w; w += al((size_t)n * 8 * 4);
    float* pooled = (float*)w; w += 256;

    const int B = 256;
    int nb_n  = (n + B - 1) / B;
    int nb_e4 = (E4 + B - 1) / B;
    if (nb_e4 < 1) nb_e4 = 1;
    int waves = (n + 15) / 16;
    int nb_t = (waves * 32 + B - 1) / B;

    // degree + symmetric normalization factors
    init_deg_kernel<<<nb_n, B, 0, stream>>>(deg, n);
    if (E4 > 0) degree4_kernel<<<nb_e4, B, 0, stream>>>(dst4, deg, E4);
    if (rem)    degree_tail_kernel<<<1, 32, 0, stream>>>(dstT, deg, rem);
    dinv_kernel<<<nb_n, B, 0, stream>>>(deg, n);

    // layer 1 (input dim 2)
    transform_wmma_kernel<2><<<nb_t, B, 0, stream>>>(x, W1, deg, t, n);
    hipMemsetAsync(agg, 0, (size_t)n * 8 * 4, stream);
    if (E4 > 0) scatter4_kernel<<<nb_e4, B, 0, stream>>>(src4, dst4, t, agg, E4);
    if (rem)    scatter_tail_kernel<<<1, 32, 0, stream>>>(srcT, dstT, t, agg, rem);
    finalize_kernel<<<nb_n, B, 0, stream>>>(t, agg, deg, b1, h, pooled, n, 0);

    // layer 2
    transform_wmma_kernel<8><<<nb_t, B, 0, stream>>>(h, W2, deg, t, n);
    hipMemsetAsync(agg, 0, (size_t)n * 8 * 4, stream);
    if (E4 > 0) scatter4_kernel<<<nb_e4, B, 0, stream>>>(src4, dst4, t, agg, E4);
    if (rem)    scatter_tail_kernel<<<1, 32, 0, stream>>>(srcT, dstT, t, agg, rem);
    finalize_kernel<<<nb_n, B, 0, stream>>>(t, agg, deg, b2, h, pooled, n, 0);

    // layer 3 + pooling
    transform_wmma_kernel<8><<<nb_t, B, 0, stream>>>(h, W3, deg, t, n);
    hipMemsetAsync(agg, 0, (size_t)n * 8 * 4, stream);
    hipMemsetAsync(pooled, 0, 8 * 4, stream);
    if (E4 > 0) scatter4_kernel<<<nb_e4, B, 0, stream>>>(src4, dst4, t, agg, E4);
    if (rem)    scatter_tail_kernel<<<1, 32, 0, stream>>>(srcT, dstT, t, agg, rem);
    finalize_kernel<<<nb_n, B, 0, stream>>>(t, agg, deg, b3, h, pooled, n, 1);

    // head
    head_kernel<<<1, 32, 0, stream>>>(pooled, Wr1, br1, Wr2, br2, (float*)d_out, n);
}